// Attention_based_Adjacency_Matrix_5334349382039
// MI455X (gfx1250) — compile-verified
//
#include <hip/hip_runtime.h>
#include <hip/hip_bf16.h>

typedef __attribute__((ext_vector_type(2))) float v2f;
typedef __attribute__((ext_vector_type(8))) float v8f;

#define TM 64
#define TN 64
#define TK 32
#define LDS_PAD 4
#define ALPHA_C 1.0f

// ---------------------------------------------------------------------------
// K0: zero the rowsum scratch (ws is poisoned / holds stale sums otherwise)
// ---------------------------------------------------------------------------
__global__ void zero_kernel(float* __restrict__ p, int n) {
    int t = blockIdx.x * blockDim.x + threadIdx.x;
    if (t < n) p[t] = 0.0f;
}

// ---------------------------------------------------------------------------
// K1: adjacency[i,j] = exp(-relu(sum_d |F[i,d]-F[j,d]| * a[d]))
// 64x64 output tile per block, 256 threads, 4x4 micro-tile per thread.
// Feature chunks staged TRANSPOSED in LDS (At[k][i]) so the inner-loop
// ds_load_b128 reads across tx are bank-conflict-free.
// ---------------------------------------------------------------------------
__global__ __launch_bounds__(256) void adj_kernel(const float* __restrict__ F,
                                                  const float* __restrict__ a,
                                                  float* __restrict__ adj,
                                                  int n, int d) {
    __shared__ alignas(16) float At[TK][TM + LDS_PAD];
    __shared__ alignas(16) float Bt[TK][TN + LDS_PAD];
    __shared__ float a_s[256];

    const int tid = threadIdx.x;
    const int tx = tid & 15;      // 0..15 -> column group
    const int ty = tid >> 4;      // 0..15 -> row group
    const int row0 = blockIdx.y * TM;
    const int col0 = blockIdx.x * TN;

    if (tid < d && tid < 256) a_s[tid] = a[tid];

    float acc[4][4];
#pragma unroll
    for (int r = 0; r < 4; ++r)
#pragma unroll
        for (int c = 0; c < 4; ++c) acc[r][c] = 0.0f;

    const int nk = d / TK;
    for (int kc = 0; kc < nk; ++kc) {
        __syncthreads();  // protect LDS from previous iteration's readers
        // Cooperative load: 64 rows x 32 cols per matrix = 512 float4, 2/thread.
#pragma unroll
        for (int q = 0; q < 2; ++q) {
            const int idx = q * 256 + tid;
            const int r  = idx >> 3;   // 0..63
            const int c4 = idx & 7;    // 0..7
            const float4 fa = *(const float4*)(F + (size_t)(row0 + r) * d + kc * TK + c4 * 4);
            const float4 fb = *(const float4*)(F + (size_t)(col0 + r) * d + kc * TK + c4 * 4);
            At[c4 * 4 + 0][r] = fa.x; At[c4 * 4 + 1][r] = fa.y;
            At[c4 * 4 + 2][r] = fa.z; At[c4 * 4 + 3][r] = fa.w;
            Bt[c4 * 4 + 0][r] = fb.x; Bt[c4 * 4 + 1][r] = fb.y;
            Bt[c4 * 4 + 2][r] = fb.z; Bt[c4 * 4 + 3][r] = fb.w;
        }
        __syncthreads();

        if (kc + 1 < nk) {
            // gfx1250 global_prefetch_b8 on next chunk
            const int r  = tid >> 3;
            const int c4 = tid & 7;
            __builtin_prefetch(F + (size_t)(row0 + r) * d + (kc + 1) * TK + c4 * 4, 0, 1);
            __builtin_prefetch(F + (size_t)(col0 + r) * d + (kc + 1) * TK + c4 * 4, 0, 1);
        }

#pragma unroll 8
        for (int k = 0; k < TK; ++k) {
            const float4 fi = *(const float4*)&At[k][ty * 4];
            const float4 fj = *(const float4*)&Bt[k][tx * 4];
            const float ak = a_s[kc * TK + k];
            const float fiv[4] = {fi.x, fi.y, fi.z, fi.w};
            const float fjv[4] = {fj.x, fj.y, fj.z, fj.w};
#pragma unroll
            for (int r = 0; r < 4; ++r)
#pragma unroll
                for (int c = 0; c < 4; ++c)
                    acc[r][c] += ak * __builtin_fabsf(fiv[r] - fjv[c]);
        }
    }

    // adjacency = exp(-relu(score)); write 4 rows x float4 (coalesced)
#pragma unroll
    for (int r = 0; r < 4; ++r) {
        float4 o;
        o.x = __expf(-ALPHA_C * fmaxf(acc[r][0], 0.0f));
        o.y = __expf(-ALPHA_C * fmaxf(acc[r][1], 0.0f));
        o.z = __expf(-ALPHA_C * fmaxf(acc[r][2], 0.0f));
        o.w = __expf(-ALPHA_C * fmaxf(acc[r][3], 0.0f));
        *(float4*)(adj + (size_t)(row0 + ty * 4 + r) * n + col0 + tx * 4) = o;
    }
}

// ---------------------------------------------------------------------------
// K2: row sums via V_WMMA_F32_16X16X4_F32 (exact f32 matmul against ones).
// A (16x4) layout per ISA: lanes 0-15 = M, VGPR0/1 = K0/K1; lanes 16-31 = K2/K3.
// D: every column N holds the row sums -> extract from lanes 0 and 16.
// One wave per (16-row strip x 1/8 column segment); atomicAdd partials.
// ---------------------------------------------------------------------------
__global__ __launch_bounds__(32) void rowsum_wmma_kernel(const float* __restrict__ adj,
                                                         float* __restrict__ rowsum,
                                                         int n) {
    const int lane = threadIdx.x;            // 0..31
    const int row  = blockIdx.x * 16 + (lane & 15);
    const int koff = (lane >> 4) * 2;        // 0 or 2
    const int segw = n >> 3;                 // columns per segment
    const int colbase = blockIdx.y * segw;

    const float2* p2 = (const float2*)(adj + (size_t)row * n + colbase + koff);

    v8f c = {};
    v2f b; b.x = 1.0f; b.y = 1.0f;

    for (int k0 = 0; k0 < segw; k0 += 4) {
        const float2 v = p2[k0 >> 1];
        v2f a2; a2.x = v.x; a2.y = v.y;      // K = koff, koff+1 for this lane
        c = __builtin_amdgcn_wmma_f32_16x16x4_f32(
                /*neg_a=*/false, a2, /*neg_b=*/false, b,
                /*c_mod=*/(short)0, c, /*reuse_a=*/false, /*reuse_b=*/false);
    }

    if ((lane & 15) == 0) {                  // lanes 0 (M=0..7) and 16 (M=8..15)
        const int rbase = blockIdx.x * 16 + ((lane >> 4) ? 8 : 0);
#pragma unroll
        for (int m = 0; m < 8; ++m) atomicAdd(&rowsum[rbase + m], c[m]);
    }
}

// ---------------------------------------------------------------------------
// K3: rowsum -> dsq = rsqrt(rowsum), in place
// ---------------------------------------------------------------------------
__global__ void rsqrt_kernel(float* __restrict__ p, int n) {
    int t = blockIdx.x * blockDim.x + threadIdx.x;
    if (t < n) p[t] = rsqrtf(p[t]);
}

// ---------------------------------------------------------------------------
// K4: normalized[i,j] = dsq[i] * adj[i,j] * dsq[j]   (streaming, float4)
// grid = (n/1024, n); blockIdx.y = row
// ---------------------------------------------------------------------------
__global__ __launch_bounds__(256) void normalize_kernel(const float* __restrict__ adj,
                                                        const float* __restrict__ dsq,
                                                        float* __restrict__ out,
                                                        int n) {
    const int i = blockIdx.y;
    const int j = (blockIdx.x * 256 + threadIdx.x) * 4;
    const size_t base = (size_t)i * n + j;
    const float4 v  = *(const float4*)(adj + base);
    const float4 dj = *(const float4*)(dsq + j);
    const float di = dsq[i];
    float4 o;
    o.x = di * v.x * dj.x;
    o.y = di * v.y * dj.y;
    o.z = di * v.z * dj.z;
    o.w = di * v.w * dj.w;
    *(float4*)(out + base) = o;
}

// ---------------------------------------------------------------------------
extern "C" void kernel_launch(void* const* d_in, const int* in_sizes, int n_in,
                              void* d_out, int out_size, void* d_ws, size_t ws_size,
                              hipStream_t stream) {
    const float* F = (const float*)d_in[0];   // (n, d) f32
    const float* a = (const float*)d_in[1];   // (d, 1) f32
    const int d = in_sizes[1];                // 256
    const int n = in_sizes[0] / d;            // 8192

    float* normalized = (float*)d_out;                         // first output
    float* adj        = (float*)d_out + (size_t)n * n;         // second output
    float* rowsum     = (float*)d_ws;                          // n floats scratch

    // K0: zero rowsum scratch (must happen every call)
    zero_kernel<<<(n + 255) / 256, 256, 0, stream>>>(rowsum, n);

    // K1: adjacency (compute-bound pairwise weighted-L1 + exp)
    dim3 g1(n / TN, n / TM);
    adj_kernel<<<g1, 256, 0, stream>>>(F, a, adj, n, d);

    // K2: row sums via WMMA f32 16x16x4 against ones
    dim3 g2(n / 16, 8);
    rowsum_wmma_kernel<<<g2, 32, 0, stream>>>(adj, rowsum, n);

    // K3: dsq = rsqrt(rowsum)
    rsqrt_kernel<<<(n + 255) / 256, 256, 0, stream>>>(rowsum, n);

    // K4: normalized = dsq_i * adj * dsq_j
    dim3 g4(n / 1024, n);
    normalize_kernel<<<g4, 256, 0, stream>>>(adj, rowsum, normalized, n);
}